// mMultiHeadSelfAttention_5884105195924
// MI455X (gfx1250) — compile-verified
//
#include <hip/hip_runtime.h>
#include <hip/hip_bf16.h>

// ---------------------------------------------------------------------------
// Multi-head self-attention forward, fp32, flash-style, CDNA5.
// WMMA f32 16x16x4 matrix pipe + async global->LDS double buffering (ASYNCcnt).
// B=4,S=4,L=1024,F=256 ; D=256,H=8,DH=32 ; N = B*S = 16 sequences.
// ---------------------------------------------------------------------------

constexpr int NL   = 16384;   // (B*S)*L rows
constexpr int FD   = 256;     // F == D == 256
constexpr int LSEQ = 1024;
constexpr int DHD  = 32;      // head dim
constexpr int NT   = LSEQ / 16;  // 64 k-tiles

typedef __attribute__((ext_vector_type(2))) float v2f;
typedef __attribute__((ext_vector_type(8))) float v8f;

__device__ __forceinline__ v8f wmma4(v2f a, v2f b, v8f c) {
  // V_WMMA_F32_16X16X4_F32 : D = A(16x4) x B(4x16) + C(16x16), exact fp32
  return __builtin_amdgcn_wmma_f32_16x16x4_f32(
      /*neg_a=*/false, a, /*neg_b=*/false, b,
      /*c_mod=*/(short)0, c, /*reuse_a=*/false, /*reuse_b=*/false);
}

// Async copy of one 16x32-float tile (row stride FD floats in global) into a
// row-major 16x32 LDS tile. 32 lanes x 16B x 4 issues = 2KB. Tracked by
// ASYNCcnt; caller waits with s_wait_asynccnt.
__device__ __forceinline__ void async_tile_copy(const float* gRowBase,
                                                unsigned ldsBase, int lane) {
  const int rsub = lane >> 3;          // 0..3 : row within group of 4
  const int cflt = (lane & 7) * 4;     // float offset of this lane's 16B chunk
  #pragma unroll
  for (int i = 0; i < 4; ++i) {
    const int row = i * 4 + rsub;
    const float* g = gRowBase + (size_t)row * FD + cflt;
    unsigned l = ldsBase + (unsigned)(row * 128 + (lane & 7) * 16);
    asm volatile("global_load_async_to_lds_b128 %0, %1, off"
                 :: "v"(l), "v"(g) : "memory");
  }
}

// ---------------------------------------------------------------------------
// C[M,N] = A[M,K] * B[N,K]^T + bias[N]    with M=NL, N=K=256.
// One wave computes a 32x64 strip: 8 accumulators, A frags reused 4x, B 2x.
// grid = 512 blocks * 128 threads (4 waves) -> 2048 waves = 512 mTiles * 4.
// ---------------------------------------------------------------------------
__global__ void __launch_bounds__(128)
gemm_abt_bias_kernel(const float* __restrict__ A, const float* __restrict__ Bm,
                     const float* __restrict__ bias, float* __restrict__ C) {
  const int wid   = (blockIdx.x * blockDim.x + threadIdx.x) >> 5;
  const int lane  = threadIdx.x & 31;
  const int mTile = wid >> 2;       // 0..511 (32 rows each)
  const int g     = wid & 3;        // which 64-col group
  const int half  = lane >> 4;      // 0: K even pair, 1: K odd pair (A/B layout)
  const int l16   = lane & 15;

  const float* aRow0 = A + (size_t)(mTile * 32 + l16) * FD + 2 * half;
  const float* aRow1 = aRow0 + (size_t)16 * FD;
  const float* bRow0 = Bm + (size_t)(g * 64 +  0 + l16) * FD + 2 * half;
  const float* bRow1 = Bm + (size_t)(g * 64 + 16 + l16) * FD + 2 * half;
  const float* bRow2 = Bm + (size_t)(g * 64 + 32 + l16) * FD + 2 * half;
  const float* bRow3 = Bm + (size_t)(g * 64 + 48 + l16) * FD + 2 * half;

  v8f a00 = {}, a01 = {}, a02 = {}, a03 = {};
  v8f a10 = {}, a11 = {}, a12 = {}, a13 = {};

  #pragma unroll 4
  for (int k0 = 0; k0 < FD; k0 += 4) {
    __builtin_prefetch(aRow0 + k0 + 64, 0, 3);   // unconditional, speculative
    __builtin_prefetch(aRow1 + k0 + 64, 0, 3);
    v2f a0 = *(const v2f*)(aRow0 + k0);
    v2f a1 = *(const v2f*)(aRow1 + k0);
    v2f b0 = *(const v2f*)(bRow0 + k0);
    v2f b1 = *(const v2f*)(bRow1 + k0);
    v2f b2 = *(const v2f*)(bRow2 + k0);
    v2f b3 = *(const v2f*)(bRow3 + k0);
    a00 = wmma4(a0, b0, a00);
    a01 = wmma4(a0, b1, a01);
    a02 = wmma4(a0, b2, a02);
    a03 = wmma4(a0, b3, a03);
    a10 = wmma4(a1, b0, a10);
    a11 = wmma4(a1, b1, a11);
    a12 = wmma4(a1, b2, a12);
    a13 = wmma4(a1, b3, a13);
  }

  // C/D layout: vgpr r, lanes0-15 -> (M=r, N=l16); lanes16-31 -> (M=r+8, N=l16)
  v8f acc[2][4] = {{a00, a01, a02, a03}, {a10, a11, a12, a13}};
  #pragma unroll
  for (int t = 0; t < 4; ++t) {
    const int col = g * 64 + t * 16 + l16;
    const float bv = bias[col];
    #pragma unroll
    for (int u = 0; u < 2; ++u) {
      #pragma unroll
      for (int r = 0; r < 8; ++r) {
        const int row = mTile * 32 + u * 16 + r + 8 * half;
        C[(size_t)row * FD + col] = acc[u][t][r] + bv;
      }
    }
  }
}

// ---------------------------------------------------------------------------
// Flash attention: one wave per (seq n, head h, 16-row q-tile).
// K/V tiles are double-buffered in LDS via GLOBAL_LOAD_ASYNC_TO_LDS_B128
// (ASYNCcnt pipeline); S=QK^T and O+=PV run on the f32 WMMA pipe; probs are
// never materialized (online softmax).
// grid = 2048 blocks * 128 threads (4 waves) -> 8192 waves = 16*8*64.
// ---------------------------------------------------------------------------
__global__ void __launch_bounds__(128)
attention_kernel(const float* __restrict__ Q, const float* __restrict__ K,
                 const float* __restrict__ V, float* __restrict__ O) {
  __shared__ float kbuf[4][2][16 * 32];     // per-wave double-buffered K tile
  __shared__ float vbuf[4][2][16 * 32];     // per-wave double-buffered V tile
  __shared__ float pbuf[4][16 * 17];        // per-wave P staging (pad 17)

  const int w    = threadIdx.x >> 5;
  const int wid  = blockIdx.x * 4 + w;
  const int lane = threadIdx.x & 31;
  const int half = lane >> 4;
  const int l16  = lane & 15;

  const int qTile = wid & 63;          // 64 q tiles of 16 rows
  const int h     = (wid >> 6) & 7;    // head
  const int n     = wid >> 9;          // sequence 0..15

  const size_t seqBase = (size_t)n * LSEQ;   // row base in [NL,256]
  const int    off     = h * DHD;            // column base of this head

  const unsigned kaddr[2] = {(unsigned)(uintptr_t)&kbuf[w][0][0],
                             (unsigned)(uintptr_t)&kbuf[w][1][0]};
  const unsigned vaddr[2] = {(unsigned)(uintptr_t)&vbuf[w][0][0],
                             (unsigned)(uintptr_t)&vbuf[w][1][0]};

  // Preload Q as A-fragments: 8 K-chunks of 4 covering DH=32.
  v2f aq[8];
  {
    const float* qRow = Q + (seqBase + qTile * 16 + l16) * FD + off + 2 * half;
    #pragma unroll
    for (int s = 0; s < 8; ++s) aq[s] = *(const v2f*)(qRow + 4 * s);
  }

  float mrow[8], lrow[8];
  #pragma unroll
  for (int r = 0; r < 8; ++r) { mrow[r] = -3.402823466e38f; lrow[r] = 0.0f; }
  v8f o0 = {}, o1 = {};                       // 16x32 output accumulator

  float* pl = &pbuf[w][0];
  const float sc = 0.17677669529663687f;      // 1/sqrt(32)

  // ---- prologue: async-stage tile 0 into buffer 0 -------------------------
  async_tile_copy(K + seqBase * FD + off, kaddr[0], lane);
  async_tile_copy(V + seqBase * FD + off, vaddr[0], lane);

  for (int kt = 0; kt < NT; ++kt) {
    // ---- issue async copies for the next tile into the other buffer ------
    const int ktn = (kt + 1 < NT) ? kt + 1 : NT - 1;   // tail: redundant reload
    const int nb  = (kt + 1) & 1;
    async_tile_copy(K + (seqBase + (size_t)ktn * 16) * FD + off, kaddr[nb], lane);
    async_tile_copy(V + (seqBase + (size_t)ktn * 16) * FD + off, vaddr[nb], lane);

    // Wait until the current tile's 8 async ops are done (8 newer remain).
    asm volatile("s_wait_asynccnt 0x8" ::: "memory");

    const float* kb = &kbuf[w][kt & 1][0];
    const float* vb = &vbuf[w][kt & 1][0];

    // ---- S = Q * K^T  (16x16, K-dim 32) : 8 WMMAs, B frags from LDS -------
    v8f S = {};
    #pragma unroll
    for (int s = 0; s < 8; ++s) {
      v2f bk = *(const v2f*)(kb + l16 * 32 + 4 * s + 2 * half);
      S = wmma4(aq[s], bk, S);
    }

    // ---- online softmax (rows live across 16-lane halves) -----------------
    float p[8];
    #pragma unroll
    for (int r = 0; r < 8; ++r) {
      float v = S[r] * sc;
      float t = v;
      t = fmaxf(t, __shfl_xor(t, 1, 32));
      t = fmaxf(t, __shfl_xor(t, 2, 32));
      t = fmaxf(t, __shfl_xor(t, 4, 32));
      t = fmaxf(t, __shfl_xor(t, 8, 32));
      const float mnew  = fmaxf(mrow[r], t);
      const float alpha = __expf(mrow[r] - mnew);
      const float pe    = __expf(v - mnew);
      float rs = pe;
      rs += __shfl_xor(rs, 1, 32);
      rs += __shfl_xor(rs, 2, 32);
      rs += __shfl_xor(rs, 4, 32);
      rs += __shfl_xor(rs, 8, 32);
      lrow[r] = lrow[r] * alpha + rs;
      mrow[r] = mnew;
      p[r]    = pe;
      o0[r] *= alpha;
      o1[r] *= alpha;
    }

    // ---- stage P (D-layout) to LDS, re-read as A-fragments ----------------
    #pragma unroll
    for (int r = 0; r < 8; ++r)
      pl[(r + 8 * half) * 17 + l16] = p[r];
    // same-wave DS ops are in-order; compiler inserts s_wait_dscnt before use

    // ---- O += P * V  (16x32, K-dim 16) : 8 WMMAs, V frags from LDS --------
    #pragma unroll
    for (int s = 0; s < 4; ++s) {
      v2f ap;
      ap.x = pl[l16 * 17 + 4 * s + 2 * half];
      ap.y = pl[l16 * 17 + 4 * s + 2 * half + 1];
      const int kr = 4 * s + 2 * half;          // V rows kr, kr+1 of this tile
      v2f b0 = { vb[kr * 32 + l16],      vb[(kr + 1) * 32 + l16]      };
      v2f b1 = { vb[kr * 32 + 16 + l16], vb[(kr + 1) * 32 + 16 + l16] };
      o0 = wmma4(ap, b0, o0);
      o1 = wmma4(ap, b1, o1);
    }
  }

  // ---- normalize and store ------------------------------------------------
  #pragma unroll
  for (int r = 0; r < 8; ++r) {
    const float inv = 1.0f / lrow[r];
    const size_t row = seqBase + qTile * 16 + r + 8 * half;
    O[row * FD + off + l16]      = o0[r] * inv;
    O[row * FD + off + 16 + l16] = o1[r] * inv;
  }
}

// ---------------------------------------------------------------------------
// Softmax pooling over L: out[n,d] = sum_l x*softmax_l(x).
// One block per sequence; thread d owns column d -> fully coalesced rows.
// ---------------------------------------------------------------------------
__global__ void __launch_bounds__(256)
pool_kernel(const float* __restrict__ X, float* __restrict__ out) {
  const int n = blockIdx.x;        // 0..15
  const int d = threadIdx.x;       // 0..255
  const float* col = X + (size_t)n * LSEQ * FD + d;
  float mx = -3.402823466e38f;
  for (int l = 0; l < LSEQ; ++l) mx = fmaxf(mx, col[(size_t)l * FD]);
  float s1 = 0.0f, s2 = 0.0f;
  for (int l = 0; l < LSEQ; ++l) {
    const float v = col[(size_t)l * FD];
    const float e = __expf(v - mx);
    s1 += e;
    s2 += v * e;
  }
  out[n * FD + d] = s2 / s1;
}

// ---------------------------------------------------------------------------
extern "C" void kernel_launch(void* const* d_in, const int* in_sizes, int n_in,
                              void* d_out, int out_size, void* d_ws, size_t ws_size,
                              hipStream_t stream) {
  const float* x  = (const float*)d_in[0];
  const float* Wq = (const float*)d_in[1];
  const float* bq = (const float*)d_in[2];
  const float* Wk = (const float*)d_in[3];
  const float* bk = (const float*)d_in[4];
  const float* Wv = (const float*)d_in[5];
  const float* bv = (const float*)d_in[6];
  const float* Wo = (const float*)d_in[7];
  const float* bo = (const float*)d_in[8];

  const size_t mat = (size_t)NL * FD;        // 16384*256 floats = 16.8 MB
  float* q    = (float*)d_ws;
  float* k    = q + mat;
  float* v    = k + mat;
  float* attn = v + mat;
  float* proj = attn + mat;                  // total 5*mat = 84 MB of ws

  // QKV projections (WMMA f32 GEMMs, 32x64 strip per wave)
  gemm_abt_bias_kernel<<<512, 128, 0, stream>>>(x, Wq, bq, q);
  gemm_abt_bias_kernel<<<512, 128, 0, stream>>>(x, Wk, bk, k);
  gemm_abt_bias_kernel<<<512, 128, 0, stream>>>(x, Wv, bv, v);

  // Flash attention: async-LDS double-buffered K/V, probs never materialized
  attention_kernel<<<2048, 128, 0, stream>>>(q, k, v, attn);

  // Output projection
  gemm_abt_bias_kernel<<<512, 128, 0, stream>>>(attn, Wo, bo, proj);

  // Softmax pooling over sequence length
  pool_kernel<<<16, 256, 0, stream>>>(proj, (float*)d_out);
}